// Memory_6047313953526
// MI455X (gfx1250) — compile-verified
//
#include <hip/hip_runtime.h>
#include <math.h>

typedef __attribute__((ext_vector_type(2))) float v2f;
typedef __attribute__((ext_vector_type(8))) float v8f;

#define TEMP     0.07f
#define MOMENTUM 0.2f
#define KTOP     1024

// ---------------------------------------------------------------------------
// Order-preserving float <-> uint key (ascending uint order == ascending float)
// ---------------------------------------------------------------------------
__device__ __forceinline__ unsigned fkey(float x) {
    unsigned u = __float_as_uint(x);
    return (u & 0x80000000u) ? ~u : (u | 0x80000000u);
}
__device__ __forceinline__ float fkeyinv(unsigned k) {
    unsigned u = (k & 0x80000000u) ? (k ^ 0x80000000u) : ~k;
    return __uint_as_float(u);
}

// ---------------------------------------------------------------------------
// mat[B,N] = f[B,D] @ features[N,D]^T using V_WMMA_F32_16X16X4_F32.
// One wave computes a 16(M) x 64(N) tile; A fragment reused 4x.
// Software-pipelined: next K-step's 5 b64 fragments are issued before the
// current step's 4 WMMAs, so the wait ahead of a WMMA group is
// s_wait_loadcnt <= 5 instead of a full drain.
// ---------------------------------------------------------------------------
__global__ void gemm_wmma(const float* __restrict__ f,
                          const float* __restrict__ feat,
                          float* __restrict__ mat,
                          int N, int D) {
    const int lane = threadIdx.x & 31;
    const int half = lane >> 4;      // 0: K+{0,1}, 1: K+{2,3}
    const int r    = lane & 15;      // M index for A, N index for B
    const int m0   = blockIdx.y * 16;
    const int n0   = blockIdx.x * 64;

    const float* pa  = f    + (size_t)(m0 + r) * D + 2 * half;
    const float* pb0 = feat + (size_t)(n0 +  0 + r) * D + 2 * half;
    const float* pb1 = feat + (size_t)(n0 + 16 + r) * D + 2 * half;
    const float* pb2 = feat + (size_t)(n0 + 32 + r) * D + 2 * half;
    const float* pb3 = feat + (size_t)(n0 + 48 + r) * D + 2 * half;

    // pipeline prologue: fragments for k = 0
    v2f ca  = *(const v2f*)(pa);
    v2f cb0 = *(const v2f*)(pb0);
    v2f cb1 = *(const v2f*)(pb1);
    v2f cb2 = *(const v2f*)(pb2);
    v2f cb3 = *(const v2f*)(pb3);

    v8f acc0 = {}, acc1 = {}, acc2 = {}, acc3 = {};

    #pragma unroll 4
    for (int k = 0; k < D; k += 4) {
        // clamped prefetch address: branch-free, last iter reloads k (unused)
        const int kn = (k + 4 < D) ? (k + 4) : k;
        v2f na  = *(const v2f*)(pa  + kn);
        v2f nb0 = *(const v2f*)(pb0 + kn);
        v2f nb1 = *(const v2f*)(pb1 + kn);
        v2f nb2 = *(const v2f*)(pb2 + kn);
        v2f nb3 = *(const v2f*)(pb3 + kn);

        acc0 = __builtin_amdgcn_wmma_f32_16x16x4_f32(false, ca, false, cb0,
                                                     (short)0, acc0, false, false);
        acc1 = __builtin_amdgcn_wmma_f32_16x16x4_f32(false, ca, false, cb1,
                                                     (short)0, acc1, false, false);
        acc2 = __builtin_amdgcn_wmma_f32_16x16x4_f32(false, ca, false, cb2,
                                                     (short)0, acc2, false, false);
        acc3 = __builtin_amdgcn_wmma_f32_16x16x4_f32(false, ca, false, cb3,
                                                     (short)0, acc3, false, false);

        ca = na; cb0 = nb0; cb1 = nb1; cb2 = nb2; cb3 = nb3;
    }

    // C/D layout: VGPR rr -> row M = rr + 8*half, col = lane&15.
    #pragma unroll
    for (int rr = 0; rr < 8; ++rr) {
        size_t rowoff = (size_t)(m0 + rr + 8 * half) * N + n0 + r;
        mat[rowoff +  0] = acc0[rr];
        mat[rowoff + 16] = acc1[rr];
        mat[rowoff + 32] = acc2[rr];
        mat[rowoff + 48] = acc3[rr];
    }
}

// ---------------------------------------------------------------------------
// Per-row: pos_min, top-K negative exp-sum via 4-pass radix select, loss.
// One 256-thread block per row; mat row (64KB) stays hot in L2.
// ---------------------------------------------------------------------------
__global__ void row_loss_kernel(const float* __restrict__ mat,
                                const int* __restrict__ labels,
                                const int* __restrict__ indexes,
                                float* __restrict__ row_loss,
                                int N) {
    __shared__ unsigned hist[256];
    __shared__ float    redf[256];
    __shared__ unsigned s_prefix, s_mask;
    __shared__ int      s_remaining;
    __shared__ float    s_posmin, s_negmax;

    const int b   = blockIdx.x;
    const int tid = threadIdx.x;
    const int myl = labels[indexes[b]];
    const float* row = mat + (size_t)b * N;

    // ---- scan 1: pos_min & neg_max -------------------------------------
    float pmin = INFINITY, nmax = -INFINITY;
    for (int n = tid; n < N; n += 256) {
        float v = row[n];
        if (labels[n] == myl) pmin = fminf(pmin, v);
        else                  nmax = fmaxf(nmax, v);
    }
    redf[tid] = pmin; __syncthreads();
    for (int s = 128; s > 0; s >>= 1) {
        if (tid < s) redf[tid] = fminf(redf[tid], redf[tid + s]);
        __syncthreads();
    }
    if (tid == 0) s_posmin = redf[0];
    __syncthreads();
    redf[tid] = nmax; __syncthreads();
    for (int s = 128; s > 0; s >>= 1) {
        if (tid < s) redf[tid] = fmaxf(redf[tid], redf[tid + s]);
        __syncthreads();
    }
    if (tid == 0) {
        s_negmax    = redf[0];
        s_prefix    = 0u;
        s_mask      = 0u;
        s_remaining = KTOP;
    }
    __syncthreads();

    // ---- scans 2-5: radix select K-th largest negative key -------------
    for (int pass = 0; pass < 4; ++pass) {
        const int shift = 24 - 8 * pass;
        hist[tid] = 0u;
        __syncthreads();
        const unsigned pre = s_prefix, msk = s_mask;
        for (int n = tid; n < N; n += 256) {
            if (labels[n] != myl) {
                unsigned u = fkey(row[n]);
                if ((u & msk) == pre)
                    atomicAdd(&hist[(u >> shift) & 255u], 1u);
            }
        }
        __syncthreads();
        if (tid == 0) {
            int rem = s_remaining;
            unsigned cum = 0;  // count strictly above chosen bin
            int bin = 0;
            for (int i = 255; i >= 0; --i) {
                if (cum + hist[i] >= (unsigned)rem) { bin = i; break; }
                cum += hist[i];
            }
            s_remaining = rem - (int)cum;          // ties taken at threshold
            s_prefix    = pre | ((unsigned)bin << shift);
            s_mask      = msk | (255u << shift);
        }
        __syncthreads();
    }
    const unsigned tkey = s_prefix;
    const int      rem  = s_remaining;
    const float    mval = fmaxf(s_posmin, s_negmax);   // max logit*TEMP

    // ---- scan 6: sum exp over keys strictly above threshold ------------
    float ssum = 0.0f;
    for (int n = tid; n < N; n += 256) {
        if (labels[n] != myl) {
            float v = row[n];
            if (fkey(v) > tkey) ssum += expf((v - mval) / TEMP);
        }
    }
    redf[tid] = ssum; __syncthreads();
    for (int s = 128; s > 0; s >>= 1) {
        if (tid < s) redf[tid] += redf[tid + s];
        __syncthreads();
    }
    if (tid == 0) {
        float tval = fkeyinv(tkey);
        float S    = redf[0] + (float)rem * expf((tval - mval) / TEMP);
        float p    = (s_posmin - mval) / TEMP;
        row_loss[b] = logf(expf(p) + S) - p;   // logsumexp - z0
    }
}

__global__ void finalize_loss(const float* __restrict__ row_loss,
                              float* __restrict__ out, int B) {
    __shared__ float red[64];
    int tid = threadIdx.x;
    red[tid] = (tid < B) ? row_loss[tid] : 0.0f;
    __syncthreads();
    for (int s = 32; s > 0; s >>= 1) {
        if (tid < s) red[tid] += red[tid + s];
        __syncthreads();
    }
    if (tid == 0) out[0] = red[0] / (float)B;
}

// ---------------------------------------------------------------------------
// Momentum update of 64 indexed rows (after bulk D2D copy of features).
// Last duplicate index wins -> deterministic scatter.
// ---------------------------------------------------------------------------
__global__ void update_rows(const float* __restrict__ features,
                            const float* __restrict__ f_weak,
                            const int* __restrict__ indexes,
                            float* __restrict__ out_feat,
                            int B, int D) {
    __shared__ float red[256];
    const int b   = blockIdx.x;
    const int tid = threadIdx.x;
    const int row = indexes[b];
    for (int b2 = b + 1; b2 < B; ++b2)
        if (indexes[b2] == row) return;        // uniform across block

    const float* src = features + (size_t)row * D;
    const float* fw  = f_weak   + (size_t)b   * D;

    float ss = 0.0f;
    for (int d = tid; d < D; d += 256) {
        float w = src[d] * MOMENTUM + fw[d] * (1.0f - MOMENTUM);
        ss += w * w;
    }
    red[tid] = ss; __syncthreads();
    for (int s = 128; s > 0; s >>= 1) {
        if (tid < s) red[tid] += red[tid + s];
        __syncthreads();
    }
    const float inv = 1.0f / fmaxf(sqrtf(red[0]), 1e-12f);
    for (int d = tid; d < D; d += 256) {
        float w = src[d] * MOMENTUM + fw[d] * (1.0f - MOMENTUM);
        out_feat[(size_t)row * D + d] = w * inv;
    }
}

// ---------------------------------------------------------------------------
extern "C" void kernel_launch(void* const* d_in, const int* in_sizes, int n_in,
                              void* d_out, int out_size, void* d_ws, size_t ws_size,
                              hipStream_t stream) {
    const float* f        = (const float*)d_in[0];
    const float* f_weak   = (const float*)d_in[1];
    const int*   indexes  = (const int*)d_in[2];
    const float* features = (const float*)d_in[3];
    const int*   labels   = (const int*)d_in[4];

    const int B = in_sizes[2];            // 64
    const int N = in_sizes[4];            // 16384
    const int D = in_sizes[0] / B;        // 2048

    float* out      = (float*)d_out;      // [0]=loss, [1..]=updated features
    float* mat      = (float*)d_ws;       // B*N floats (4 MB)
    float* row_loss = mat + (size_t)B * N;

    dim3 gg(N / 64, B / 16);
    gemm_wmma<<<gg, 32, 0, stream>>>(f, features, mat, N, D);
    row_loss_kernel<<<B, 256, 0, stream>>>(mat, labels, indexes, row_loss, N);
    finalize_loss<<<1, 64, 0, stream>>>(row_loss, out, B);

    hipMemcpyAsync(out + 1, features, (size_t)N * D * sizeof(float),
                   hipMemcpyDeviceToDevice, stream);
    update_rows<<<B, 256, 0, stream>>>(features, f_weak, indexes, out + 1, B, D);
}